// SConv2d_66640712564931
// MI455X (gfx1250) — compile-verified
//
#include <hip/hip_runtime.h>

// ---------------------------------------------------------------------------
// SConv2d soft-MAJ3 tree conv for MI455X (gfx1250).
//   x: (8,3,128,128) f32, w: (32,3,3,3) f32  ->  out: (8,32,126,126) f32
// maj3(a,b,c) = (a+b+c - abc)/2 (bipolar identity); level-1 cubic factorizes:
// (p0w0)(p1w1)(p2w2) = (p0p1p2)*(w0w1w2). Scale factors deferred through the
// tree (u=2m, v=4q) to drop per-level *0.5 muls. Compute-bound in FP32 VALU:
// float2 lanes (v_pk_fma_f32), async global->LDS staging, packed stores.
// ---------------------------------------------------------------------------

#define NN    8
#define CCh   3
#define HH    128
#define WWid  128
#define OCC   32
#define HO    126
#define WO    126
#define RB    4               // output rows per block
#define INRP  8               // padded input rows per channel (need 6, pad->8 for shift addressing)
#define NGRP  ((HO + RB - 1) / RB)   // 32 row groups

typedef float v2f __attribute__((ext_vector_type(2)));
typedef float v4f __attribute__((ext_vector_type(4)));
typedef int   v4i_g __attribute__((vector_size(16)));   // matches async builtin param type

#if defined(__gfx1250__) && __has_builtin(__builtin_amdgcn_global_load_async_to_lds_b128)
#define HAS_ASYNC 1
typedef __attribute__((address_space(1))) v4i_g* as1_v4i_ptr;
typedef __attribute__((address_space(3))) v4i_g* as3_v4i_ptr;
#else
#define HAS_ASYNC 0
#endif

#if __has_builtin(__builtin_elementwise_fma)
#define VFMA(a, b, c) __builtin_elementwise_fma((a), (b), (c))
#else
#define VFMA(a, b, c) ((a) * (b) + (c))
#endif

__device__ __forceinline__ v2f splat(float s) { v2f r; r.x = s; r.y = s; return r; }
__device__ __forceinline__ v2f mk2(float a, float b) { v2f r; r.x = a; r.y = b; return r; }

// Level-2 maj3, inputs u = 2*m, output v = 4*maj3(m0,m1,m2):
//   v = (u0+u1+u2) - 0.25*u0*u1*u2
__device__ __forceinline__ v2f maj3_s2(v2f u0, v2f u1, v2f u2) {
    v2f sum = u0 + u1 + u2;
    v2f t   = u0 * u1;
    t = t * splat(-0.25f);
    return VFMA(t, u2, sum);
}

// Level-3 maj3, inputs v = 4*q, output r = maj3(q0,q1,q2):
//   r = ((v0+v1+v2) - (v0*v1*v2)/16) / 8
__device__ __forceinline__ v2f maj3_s4_final(v2f v0, v2f v1, v2f v2_) {
    v2f sum = v0 + v1 + v2_;
    v2f t   = v0 * v1;
    v2f t2  = t * v2_;
    v2f inner = VFMA(t2, splat(-0.0625f), sum);
    return inner * splat(0.125f);
}

__global__ __launch_bounds__(256) void sconv_maj3_kernel(
    const float* __restrict__ x, const float* __restrict__ w,
    float* __restrict__ out) {
    __shared__ float sIn[CCh * INRP * WWid];  // 12 KB padded input tile
    __shared__ float sW[OCC * 27];            // weights
    __shared__ float sWT[OCC * 9];            // per-oc triple products w0*w1*w2

    const int blk = blockIdx.x;
    const int n   = blk / NGRP;
    const int g   = blk % NGRP;
    const int r0  = g * RB;
    const int tid = threadIdx.y * 64 + threadIdx.x;

    // ---- Stage input tile: 3 ch x 8 rows x 32 segs = 768 x 16B transfers,
    //      exactly 3 per thread, shift/mask indexing only (no div/mod). ----
#pragma unroll
    for (int it = 0; it < 3; ++it) {
        const int i   = tid + it * 256;
        const int seg = i & 31;          // 16B segment within 128-col line
        const int rr  = (i >> 5) & 7;    // padded row 0..7 (rows 6,7 unused)
        const int ch  = i >> 8;          // 0..2
        int srcRow = r0 + rr;
        if (srcRow > HH - 1) srcRow = HH - 1;   // clamp tail rows (unused by compute)
        const float* src = x + (((n * CCh + ch) * HH + srcRow) * WWid + seg * 4);
        float*       dst = &sIn[((ch << 3) + rr) * WWid + seg * 4];
#if HAS_ASYNC
        __builtin_amdgcn_global_load_async_to_lds_b128(
            (as1_v4i_ptr)src, (as3_v4i_ptr)dst, 0, 0);
#else
        *(v4f*)dst = *(const v4f*)src;
#endif
    }

    // ---- Stage weights + per-oc triple products ----
    if (tid < OCC) {
        const float* wp = w + tid * 27;
#pragma unroll
        for (int k = 0; k < 27; ++k) sW[tid * 27 + k] = wp[k];
#pragma unroll
        for (int j = 0; j < 9; ++j)
            sWT[tid * 9 + j] = wp[3 * j] * wp[3 * j + 1] * wp[3 * j + 2];
    }

#if HAS_ASYNC
#if __has_builtin(__builtin_amdgcn_s_wait_asynccnt)
    __builtin_amdgcn_s_wait_asynccnt(0);
#else
    asm volatile("s_wait_asynccnt 0" ::: "memory");
#endif
#endif
    __syncthreads();

    const int t   = threadIdx.x;   // column-pair index: cols (2t, 2t+1)
    const int y   = threadIdx.y;   // row within group
    const int row = r0 + y;
    if (t >= WO / 2 || row >= HO) return;   // no further barriers below

    // ---- Gather 27 patch values for both columns (kept in VGPRs, reused 32x) ----
    v2f p[27];
#pragma unroll
    for (int ch = 0; ch < 3; ++ch) {
#pragma unroll
        for (int dy = 0; dy < 3; ++dy) {
            const float* line = &sIn[((ch << 3) + (y + dy)) * WWid + 2 * t];
            v2f a = *(const v2f*)(line);       // cols 2t, 2t+1   (8B aligned)
            v2f b = *(const v2f*)(line + 2);   // cols 2t+2, 2t+3 (8B aligned)
            const int base = (ch * 3 + dy) * 3;
            p[base + 0] = a;                   // dx = 0
            p[base + 1] = mk2(a.y, b.x);       // dx = 1
            p[base + 2] = b;                   // dx = 2
        }
    }

    // oc-independent patch triple products
    v2f P[9];
#pragma unroll
    for (int j = 0; j < 9; ++j) P[j] = p[3 * j] * p[3 * j + 1] * p[3 * j + 2];

    // ---- Loop over output channels: 57 packed VALU ops per iteration ----
#pragma unroll 4
    for (int oc = 0; oc < OCC; ++oc) {
        const float* wk = &sW[oc * 27];   // uniform (broadcast) LDS reads
        const float* wt = &sWT[oc * 9];

        // level-1: u_j = (p0w0 + p1w1 + p2w2) - P_j*W_j   (= 2*m_j)
        v2f u[9];
#pragma unroll
        for (int j = 0; j < 9; ++j) {
            v2f s = p[3 * j] * splat(wk[3 * j]);
            s = VFMA(p[3 * j + 1], splat(wk[3 * j + 1]), s);
            s = VFMA(p[3 * j + 2], splat(wk[3 * j + 2]), s);
            u[j] = VFMA(P[j], splat(-wt[j]), s);
        }
        // level-2 (inputs 2m -> outputs 4q), level-3 (inputs 4q -> final)
        v2f v0 = maj3_s2(u[0], u[1], u[2]);
        v2f v1 = maj3_s2(u[3], u[4], u[5]);
        v2f v2_ = maj3_s2(u[6], u[7], u[8]);
        v2f r  = maj3_s4_final(v0, v1, v2_);

        float* o = out + (((size_t)(n * OCC + oc) * HO + row) * WO + 2 * t);
        *(v2f*)o = r;   // 8B aligned (WO even, 2t even), coalesced
    }
}

extern "C" void kernel_launch(void* const* d_in, const int* in_sizes, int n_in,
                              void* d_out, int out_size, void* d_ws, size_t ws_size,
                              hipStream_t stream) {
    const float* x = (const float*)d_in[0];
    const float* w = (const float*)d_in[1];
    float* out = (float*)d_out;
    (void)in_sizes; (void)n_in; (void)out_size; (void)d_ws; (void)ws_size;

    dim3 grid(NN * NGRP);   // 8 images x 32 row-groups = 256 blocks
    dim3 block(64, RB);     // 64 column-pair threads x 4 rows = 8 waves
    hipLaunchKernelGGL(sconv_maj3_kernel, grid, block, 0, stream, x, w, out);
}